// VICRegLocalLoss_33483565040285
// MI455X (gfx1250) — compile-verified
//
#include <hip/hip_runtime.h>
#include <cstdint>

// Problem constants (match reference)
constexpr int kB  = 4;
constexpr int kC  = 256;
constexpr int kHW = 4096;          // 64*64
constexpr int kG  = 10;            // GAMMA / top-k
constexpr float kEPS = 1.0e-4f;

typedef float    v2f __attribute__((ext_vector_type(2)));
typedef float    v8f __attribute__((ext_vector_type(8)));
typedef uint32_t v4u __attribute__((ext_vector_type(4)));
typedef int      v4i __attribute__((ext_vector_type(4)));
typedef int      v8i __attribute__((ext_vector_type(8)));

// Total order used by jax.lax.top_k: larger value first, ties -> lower index.
__device__ __forceinline__ bool better(float v1, uint32_t i1, float v2, uint32_t i2) {
    return (v1 > v2) || (v1 == v2 && i1 < i2);
}

__device__ __forceinline__ float block_reduce_sum(float v, float* red) {
    int t = threadIdx.x;
    red[t] = v; __syncthreads();
    for (int off = 128; off > 0; off >>= 1) {
        if (t < off) red[t] += red[t + off];
        __syncthreads();
    }
    float r = red[0]; __syncthreads();
    return r;
}

// LDS byte offset of a shared-memory pointer (flat LDS aperture: offset = addr[31:0])
__device__ __forceinline__ uint32_t lds_off(const void* p) {
    return (uint32_t)(uintptr_t)p;
}

// ---------------------------------------------------------------------------
// TDM: load `rows` x 64 f32 tile (global row stride 4096 elems) into LDS with
// 16 pad dwords after every 64 dwords -> LDS row stride 80 dwords (bank-shift
// 16/row => conflict-free k-major fragment reads).  D# per ISA §8.3/§8.4.
// ---------------------------------------------------------------------------
__device__ __forceinline__ void tdm_load_rows(const float* gsrc, uint32_t lds_byte,
                                              uint32_t rows) {
    uint64_t ga = (uint64_t)(uintptr_t)gsrc;
    v4u g0;
    g0[0] = 1u;                                          // count=1, user mode
    g0[1] = lds_byte;                                    // lds_addr
    g0[2] = (uint32_t)ga;                                // global_addr[31:0]
    g0[3] = (uint32_t)((ga >> 32) & 0x01FFFFFFu)         // global_addr[56:32]
          | 0x80000000u;                                 // type=2 ("image")
    v8i g1;
    g1[0] = (int)((2u << 16)      // data_size = 4 bytes
                | (1u << 20)      // pad_enable
                | (5u << 22)      // pad_interval: 64 dwords
                | (15u << 25));   // pad_amount: 16 dwords
    g1[1] = (int)(4096u << 16);                          // tensor_dim0 lo16
    g1[2] = (int)((4096u >> 16) | (4096u << 16));        // dim0 hi16 | dim1 lo16
    g1[3] = (int)((4096u >> 16) | (64u << 16));          // dim1 hi16 | tile_dim0=64
    g1[4] = (int)rows;                                   // tile_dim1 | tile_dim2=0
    g1[5] = (int)4096u;                                  // tensor_dim0_stride lo32
    g1[6] = 0;                                           // stride0 hi16 | stride1 lo16
    g1[7] = 0;
    v4i z4 = {0, 0, 0, 0};
    v8i z8 = {0, 0, 0, 0, 0, 0, 0, 0};
    __builtin_amdgcn_tensor_load_to_lds(g0, g1, z4, z4, z8, 0);
}

// ---------------------------------------------------------------------------
// 1) Spatial means: g[t][b][c] = mean over HW.  grid = 2*B*C = 2048 blocks.
// ---------------------------------------------------------------------------
__global__ void means_kernel(const float* __restrict__ X1,
                             const float* __restrict__ X2,
                             float* __restrict__ g) {
    int bid = blockIdx.x;
    int t   = bid >> 10;
    int bc  = bid & 1023;          // b*256 + c
    const float* row = (t ? X2 : X1) + (size_t)bc * kHW;
    float s = 0.f;
    for (int i = threadIdx.x; i < kHW; i += 256) s += row[i];
    __shared__ float red[256];
    s = block_reduce_sum(s, red);
    if (threadIdx.x == 0) g[t * 1024 + bc] = s * (1.0f / kHW);
}

// ---------------------------------------------------------------------------
// 2) Feature-row squared norms: norms[t][b][pix] = sum_c X[b,c,pix]^2.
// ---------------------------------------------------------------------------
__global__ void norms_kernel(const float* __restrict__ X1,
                             const float* __restrict__ X2,
                             float* __restrict__ norms) {
    int gidx = blockIdx.x * 256 + threadIdx.x;   // 0..32767
    int t    = gidx >> 14;
    int bp   = gidx & 16383;
    int b    = bp >> 12;
    int pix  = bp & 4095;
    const float* base = (t ? X2 : X1) + (size_t)b * kC * kHW + pix;
    float s = 0.f;
    for (int c = 0; c < kC; ++c) {
        float v = base[(size_t)c * kHW];
        s = fmaf(v, v, s);
    }
    norms[gidx] = s;
}

// ---------------------------------------------------------------------------
// 3) Global VICReg on g1,g2 (4 x 256).  One block. Writes d_out[0..2].
// ---------------------------------------------------------------------------
__global__ void global_vicreg_kernel(const float* __restrict__ g,
                                     float* __restrict__ out) {
    __shared__ float Xc[2][4][256];
    __shared__ float red[256];
    int c = threadIdx.x;

    float varL[2];
    for (int t = 0; t < 2; ++t) {
        float x0 = g[t * 1024 + 0 * 256 + c];
        float x1 = g[t * 1024 + 1 * 256 + c];
        float x2 = g[t * 1024 + 2 * 256 + c];
        float x3 = g[t * 1024 + 3 * 256 + c];
        float m  = 0.25f * (x0 + x1 + x2 + x3);
        float d0 = x0 - m, d1 = x1 - m, d2 = x2 - m, d3 = x3 - m;
        Xc[t][0][c] = d0; Xc[t][1][c] = d1; Xc[t][2][c] = d2; Xc[t][3][c] = d3;
        float var = (d0 * d0 + d1 * d1 + d2 * d2 + d3 * d3) * (1.0f / 3.0f);
        float sd  = sqrtf(var + kEPS);
        float r   = 1.0f - sd; if (r < 0.f) r = 0.f;
        varL[t] = block_reduce_sum(r, red) * (1.0f / 256.0f);
    }
    __syncthreads();

    float s = 0.f;
    for (int b = 0; b < 4; ++b) {
        float d = g[b * 256 + c] - g[1024 + b * 256 + c];
        s = fmaf(d, d, s);
    }
    float inv = block_reduce_sum(s, red) * (1.0f / 1024.0f);

    float covL[2];
    for (int t = 0; t < 2; ++t) {
        float col[4];
        #pragma unroll
        for (int b = 0; b < 4; ++b) col[b] = Xc[t][b][c];
        float acc = 0.f;
        for (int j = 0; j < 256; ++j) {
            float sv = 0.f;
            #pragma unroll
            for (int b = 0; b < 4; ++b) sv = fmaf(col[b], Xc[t][b][j], sv);
            sv *= (1.0f / 3.0f);
            if (j != c) acc = fmaf(sv, sv, acc);
        }
        covL[t] = block_reduce_sum(acc, red) * (1.0f / 256.0f);
    }

    if (threadIdx.x == 0) {
        out[0] = 25.0f * inv;
        out[1] = 25.0f * 0.5f * (varL[0] + varL[1]);
        out[2] = 0.1f * 0.5f * (covL[0] + covL[1]);
    }
}

// ---------------------------------------------------------------------------
// 4) "loc" path: coordinate cdist + per-block top-10.  grid = B*64 blocks.
// ---------------------------------------------------------------------------
__global__ void __launch_bounds__(256)
loc_topk_kernel(const float* __restrict__ box1, const float* __restrict__ box2,
                float* __restrict__ candV, uint32_t* __restrict__ candI) {
    int b  = blockIdx.x >> 6;
    int r0 = (blockIdx.x & 63) * 64;

    float a1x = box1[b * 4 + 0], a1y = box1[b * 4 + 1];
    float s1x = box1[b * 4 + 2] - a1x, s1y = box1[b * 4 + 3] - a1y;
    float a2x = box2[b * 4 + 0], a2y = box2[b * 4 + 1];
    float s2x = box2[b * 4 + 2] - a2x, s2y = box2[b * 4 + 3] - a2y;

    float qx[16], qy[16];
    #pragma unroll
    for (int cc = 0; cc < 16; ++cc) {
        int m = threadIdx.x * 16 + cc;
        qx[cc] = (float)(m >> 6) * s2x + a2x;
        qy[cc] = (float)(m & 63) * s2y + a2y;
    }

    float    tv[kG];
    uint32_t ti[kG];
    #pragma unroll
    for (int k = 0; k < kG; ++k) { tv[k] = -__builtin_inff(); ti[k] = 0xffffffffu; }

    for (int rr = 0; rr < 64; ++rr) {
        int n = r0 + rr;
        float px = (float)(n >> 6) * s1x + a1x;
        float py = (float)(n & 63) * s1y + a1y;
        #pragma unroll
        for (int cc = 0; cc < 16; ++cc) {
            int m = threadIdx.x * 16 + cc;
            float dx = px - qx[cc], dy = py - qy[cc];
            float d2 = fmaf(dx, dx, dy * dy);
            uint32_t idx = (uint32_t)n * 4096u + (uint32_t)m;
            if (better(d2, idx, tv[kG - 1], ti[kG - 1])) {
                tv[kG - 1] = d2; ti[kG - 1] = idx;
                #pragma unroll
                for (int k = kG - 1; k > 0; --k) {
                    if (better(tv[k], ti[k], tv[k - 1], ti[k - 1])) {
                        float fv = tv[k]; tv[k] = tv[k - 1]; tv[k - 1] = fv;
                        uint32_t fi = ti[k]; ti[k] = ti[k - 1]; ti[k - 1] = fi;
                    }
                }
            }
        }
    }

    __shared__ float    sV[256];
    __shared__ uint32_t sI[256];
    __shared__ float    pV;
    __shared__ uint32_t pI;
    if (threadIdx.x == 0) { pV = __builtin_inff(); pI = 0u; }
    __syncthreads();

    int base = blockIdx.x * kG;
    for (int r = 0; r < kG; ++r) {
        float bv = -__builtin_inff(); uint32_t bi = 0xffffffffu; bool found = false;
        float cpV = pV; uint32_t cpI = pI;
        #pragma unroll
        for (int k = 0; k < kG; ++k) {
            bool ok = !found && better(cpV, cpI, tv[k], ti[k]);
            if (ok) { bv = tv[k]; bi = ti[k]; found = true; }
        }
        sV[threadIdx.x] = bv; sI[threadIdx.x] = bi;
        __syncthreads();
        for (int off = 128; off > 0; off >>= 1) {
            if (threadIdx.x < off &&
                better(sV[threadIdx.x + off], sI[threadIdx.x + off],
                       sV[threadIdx.x], sI[threadIdx.x])) {
                sV[threadIdx.x] = sV[threadIdx.x + off];
                sI[threadIdx.x] = sI[threadIdx.x + off];
            }
            __syncthreads();
        }
        if (threadIdx.x == 0) {
            candV[base + r] = sV[0]; candI[base + r] = sI[0];
            pV = sV[0]; pI = sI[0];
        }
        __syncthreads();
    }
}

// ---------------------------------------------------------------------------
// 5) "feat" path: fp32 WMMA GEMM with TDM double-buffered LDS staging, fused
//    d^2 epilogue + per-tile top-10.  grid = (64,64,B); 256 threads = 8 waves;
//    block computes a 64x64 tile; each wave a 16x32 strip (two 16x16 accs).
//    LDS staging layout: k-major [16][80] (TDM pads 16 dw per 64 dw row).
// ---------------------------------------------------------------------------
__global__ void __launch_bounds__(256)
feat_gemm_topk_kernel(const float* __restrict__ X1, const float* __restrict__ X2,
                      const float* __restrict__ norms,
                      float* __restrict__ candV, uint32_t* __restrict__ candI) {
    int b  = blockIdx.z;
    int tM = blockIdx.x * 64;     // rows  (fX1 pixels)
    int tN = blockIdx.y * 64;     // cols  (fX2 pixels)

    __shared__ float Ab[2][16 * 80];   // double-buffered A chunk, k-major
    __shared__ float Bb[2][16 * 80];   // double-buffered B chunk
    __shared__ float Ds[64][64];       // d^2 tile
    __shared__ float n1s[64], n2s[64];

    const float* Ag = X1 + (size_t)b * kC * kHW + tM;  // + k*kHW for row k
    const float* Bg = X2 + (size_t)b * kC * kHW + tN;

    int tid = threadIdx.x;
    if (tid < 64)        n1s[tid]      = norms[b * 4096 + tM + tid];
    else if (tid < 128)  n2s[tid - 64] = norms[16384 + b * 4096 + tN + (tid - 64)];

    int wave = tid >> 5, lane = tid & 31;
    int waveM = (wave >> 1) * 16;   // 0,16,32,48
    int waveN = (wave & 1) * 32;    // 0,32
    int lm = lane & 15, lh = lane >> 4;

    // Each wave DMAs its own 2 rows of A and B per chunk (wave-uniform D#).
    int wrow = wave * 2;                          // k-row pair owned by this wave
    uint32_t aLds[2] = { lds_off(&Ab[0][0]) + (uint32_t)wrow * 320u,
                         lds_off(&Ab[1][0]) + (uint32_t)wrow * 320u };
    uint32_t bLds[2] = { lds_off(&Bb[0][0]) + (uint32_t)wrow * 320u,
                         lds_off(&Bb[1][0]) + (uint32_t)wrow * 320u };

    v8f acc0 = {0.f, 0.f, 0.f, 0.f, 0.f, 0.f, 0.f, 0.f};
    v8f acc1 = acc0;

    // Prologue: stage chunk 0 into buffer 0.
    tdm_load_rows(Ag + (size_t)wrow * kHW, aLds[0], 2u);
    tdm_load_rows(Bg + (size_t)wrow * kHW, bLds[0], 2u);

    int buf = 0;
    for (int k0 = 0; k0 < kC; k0 += 16) {
        if (k0 + 16 < kC) {
            // Stage next chunk into the other buffer (freed at end of prev iter).
            tdm_load_rows(Ag + (size_t)(k0 + 16 + wrow) * kHW, aLds[buf ^ 1], 2u);
            tdm_load_rows(Bg + (size_t)(k0 + 16 + wrow) * kHW, bLds[buf ^ 1], 2u);
            __builtin_amdgcn_s_wait_tensorcnt(2);   // oldest pair (this chunk) done
        } else {
            __builtin_amdgcn_s_wait_tensorcnt(0);
        }
        __syncthreads();                            // all waves' slices visible

        const float* Ac = &Ab[buf][0];
        const float* Bc = &Bb[buf][0];
        #pragma unroll
        for (int kk = 0; kk < 16; kk += 4) {
            int kp = kk + 2 * lh;                   // K pair for this lane half
            v2f a  = { Ac[kp * 80 + waveM + lm],      Ac[(kp + 1) * 80 + waveM + lm] };
            v2f b0 = { Bc[kp * 80 + waveN + lm],      Bc[(kp + 1) * 80 + waveN + lm] };
            v2f b1 = { Bc[kp * 80 + waveN + 16 + lm], Bc[(kp + 1) * 80 + waveN + 16 + lm] };
            acc0 = __builtin_amdgcn_wmma_f32_16x16x4_f32(
                false, a, false, b0, (short)0, acc0, false, false);
            acc1 = __builtin_amdgcn_wmma_f32_16x16x4_f32(
                false, a, false, b1, (short)0, acc1, false, false);
        }
        __syncthreads();                            // buffer free for re-staging
        buf ^= 1;
    }

    // d^2 epilogue: C-frag (m = r + 8*laneHalf, n = lane&15) per ISA layout
    #pragma unroll
    for (int r = 0; r < 8; ++r) {
        int m  = waveM + r + 8 * lh;
        int n0 = waveN + lm;
        Ds[m][n0]      = n1s[m] + n2s[n0]      - 2.0f * acc0[r];
        Ds[m][n0 + 16] = n1s[m] + n2s[n0 + 16] - 2.0f * acc1[r];
    }
    __syncthreads();

    // Tile-wide top-10 (rank by d^2; sqrt is monotone)
    __shared__ float    sV[256];
    __shared__ uint32_t sI[256];
    __shared__ float    pV;
    __shared__ uint32_t pI;
    if (tid == 0) { pV = __builtin_inff(); pI = 0u; }
    __syncthreads();

    int cbase = ((b * 4096) + blockIdx.y * 64 + blockIdx.x) * kG;
    for (int r = 0; r < kG; ++r) {
        float bv = -__builtin_inff(); uint32_t bi = 0xffffffffu;
        float cpV = pV; uint32_t cpI = pI;
        #pragma unroll
        for (int j = 0; j < 16; ++j) {
            int e = tid * 16 + j;
            int m = e >> 6, n = e & 63;
            float v = Ds[m][n];
            uint32_t idx = (uint32_t)(tM + m) * 4096u + (uint32_t)(tN + n);
            if (better(cpV, cpI, v, idx) && better(v, idx, bv, bi)) { bv = v; bi = idx; }
        }
        sV[tid] = bv; sI[tid] = bi;
        __syncthreads();
        for (int off = 128; off > 0; off >>= 1) {
            if (tid < off &&
                better(sV[tid + off], sI[tid + off], sV[tid], sI[tid])) {
                sV[tid] = sV[tid + off]; sI[tid] = sI[tid + off];
            }
            __syncthreads();
        }
        if (tid == 0) {
            candV[cbase + r] = sV[0]; candI[cbase + r] = sI[0];
            pV = sV[0]; pI = sI[0];
        }
        __syncthreads();
    }
}

// ---------------------------------------------------------------------------
// 6) Merge per-block candidates -> final top-10 rows per (path, batch).
// ---------------------------------------------------------------------------
__global__ void merge_topk_kernel(const float* __restrict__ locV, const uint32_t* __restrict__ locI,
                                  const float* __restrict__ featV, const uint32_t* __restrict__ featI,
                                  int* __restrict__ topRows) {
    int b = blockIdx.x, path = blockIdx.y;
    const float* V; const uint32_t* I; int count;
    if (path == 0) { V = locV  + b * 64 * kG;   I = locI  + b * 64 * kG;   count = 64 * kG; }
    else           { V = featV + b * 4096 * kG; I = featI + b * 4096 * kG; count = 4096 * kG; }

    __shared__ float    sV[256];
    __shared__ uint32_t sI[256];
    __shared__ float    pV;
    __shared__ uint32_t pI;
    if (threadIdx.x == 0) { pV = __builtin_inff(); pI = 0u; }
    __syncthreads();

    for (int r = 0; r < kG; ++r) {
        float bv = -__builtin_inff(); uint32_t bi = 0xffffffffu;
        float cpV = pV; uint32_t cpI = pI;
        for (int e = threadIdx.x; e < count; e += 256) {
            float v = V[e]; uint32_t i = I[e];
            if (better(cpV, cpI, v, i) && better(v, i, bv, bi)) { bv = v; bi = i; }
        }
        sV[threadIdx.x] = bv; sI[threadIdx.x] = bi;
        __syncthreads();
        for (int off = 128; off > 0; off >>= 1) {
            if (threadIdx.x < off &&
                better(sV[threadIdx.x + off], sI[threadIdx.x + off],
                       sV[threadIdx.x], sI[threadIdx.x])) {
                sV[threadIdx.x] = sV[threadIdx.x + off];
                sI[threadIdx.x] = sI[threadIdx.x + off];
            }
            __syncthreads();
        }
        if (threadIdx.x == 0) {
            pV = sV[0]; pI = sI[0];
            topRows[(path * kB + b) * kG + r] = (int)(sI[0] / 4096u);
        }
        __syncthreads();
    }
}

// ---------------------------------------------------------------------------
// 7) Local VICReg on gathered features (40 x 256) per path.  grid = 2 blocks.
// ---------------------------------------------------------------------------
__global__ void __launch_bounds__(256)
local_vicreg_kernel(const float* __restrict__ X1, const float* __restrict__ X2,
                    const int* __restrict__ topRows, float* __restrict__ pathLoss) {
    int path = blockIdx.x;
    __shared__ float S[40][257];
    __shared__ float red[256];
    __shared__ int   spix[40];
    int c = threadIdx.x;
    if (c < 40) spix[c] = topRows[path * kB * kG + c];
    __syncthreads();

    float invacc = 0.f;
    for (int j = 0; j < 40; ++j) {
        int bb = j / kG, pix = spix[j];
        float v1 = X1[(size_t)bb * kC * kHW + (size_t)c * kHW + pix];
        float v2 = X2[(size_t)bb * kC * kHW + (size_t)c * kHW + pix];
        S[j][c] = v1;
        float d = v1 - v2;
        invacc = fmaf(d, d, invacc);
    }
    __syncthreads();
    float inv = block_reduce_sum(invacc, red) * (1.0f / (40.0f * 256.0f));

    float varL[2], covL[2];
    for (int t = 0; t < 2; ++t) {
        if (t == 1) {
            __syncthreads();
            for (int j = 0; j < 40; ++j) {
                int bb = j / kG, pix = spix[j];
                S[j][c] = X2[(size_t)bb * kC * kHW + (size_t)c * kHW + pix];
            }
            __syncthreads();
        }
        float m = 0.f;
        for (int j = 0; j < 40; ++j) m += S[j][c];
        m *= (1.0f / 40.0f);
        float vs = 0.f;
        for (int j = 0; j < 40; ++j) {
            float d = S[j][c] - m;
            S[j][c] = d;
            vs = fmaf(d, d, vs);
        }
        __syncthreads();
        vs *= (1.0f / 39.0f);
        float sd = sqrtf(vs + kEPS);
        float rl = 1.0f - sd; if (rl < 0.f) rl = 0.f;
        varL[t] = block_reduce_sum(rl, red) * (1.0f / 256.0f);

        float col[40];
        #pragma unroll
        for (int k = 0; k < 40; ++k) col[k] = S[k][c];
        float acc = 0.f;
        for (int j = 0; j < 256; ++j) {
            float sdot = 0.f;
            #pragma unroll
            for (int k = 0; k < 40; ++k) sdot = fmaf(col[k], S[k][j], sdot);
            sdot *= (1.0f / 39.0f);
            if (j != c) acc = fmaf(sdot, sdot, acc);
        }
        covL[t] = block_reduce_sum(acc, red) * (1.0f / 256.0f);
    }

    if (threadIdx.x == 0) {
        float var = 0.5f * (varL[0] + varL[1]);
        float cov = (covL[0] + covL[1]) * (1.0f / 20.0f);   // /adj(=10) /2
        pathLoss[path] = (var + cov + inv) * 0.1f;          // /g
    }
}

__global__ void combine_kernel(const float* __restrict__ pathLoss, float* __restrict__ out) {
    if (threadIdx.x == 0) out[3] = pathLoss[0] + pathLoss[1];
}

// ---------------------------------------------------------------------------
extern "C" void kernel_launch(void* const* d_in, const int* in_sizes, int n_in,
                              void* d_out, int out_size, void* d_ws, size_t ws_size,
                              hipStream_t stream) {
    (void)in_sizes; (void)n_in; (void)out_size; (void)ws_size;
    const float* X1   = (const float*)d_in[0];
    const float* X2   = (const float*)d_in[1];
    const float* box1 = (const float*)d_in[2];
    const float* box2 = (const float*)d_in[3];
    float* out = (float*)d_out;
    float* ws  = (float*)d_ws;

    // workspace layout (float slots)
    float*    g        = ws;                                 // 2048
    float*    norms    = ws + 2048;                          // 32768
    float*    locV     = ws + 34816;                         // 2560
    uint32_t* locI     = (uint32_t*)(ws + 37376);            // 2560
    float*    featV    = ws + 39936;                         // 163840
    uint32_t* featI    = (uint32_t*)(ws + 203776);           // 163840
    int*      topRows  = (int*)(ws + 367616);                // 80
    float*    pathLoss = ws + 367696;                        // 2

    means_kernel<<<2048, 256, 0, stream>>>(X1, X2, g);
    norms_kernel<<<128, 256, 0, stream>>>(X1, X2, norms);
    global_vicreg_kernel<<<1, 256, 0, stream>>>(g, out);
    loc_topk_kernel<<<256, 256, 0, stream>>>(box1, box2, locV, locI);
    feat_gemm_topk_kernel<<<dim3(64, 64, kB), 256, 0, stream>>>(X1, X2, norms, featV, featI);
    merge_topk_kernel<<<dim3(kB, 2), 256, 0, stream>>>(locV, locI, featV, featI, topRows);
    local_vicreg_kernel<<<2, 256, 0, stream>>>(X1, X2, topRows, pathLoss);
    combine_kernel<<<1, 1, 0, stream>>>(pathLoss, out);
}